// Model11_85598698209833
// MI455X (gfx1250) — compile-verified
//
#include <hip/hip_runtime.h>
#include <hip/hip_bf16.h>
#include <stdint.h>

// Problem shape (fixed by the reference's setup_inputs()).
constexpr int N  = 4;
constexpr int C  = 32;
constexpr int H  = 512;
constexpr int W  = 512;
constexpr int GH = 512;
constexpr int GW = 512;
constexpr int HW = H * W;
constexpr int PTS = N * GH * GW;        // 1,048,576 grid points
constexpr int BLOCK = 256;              // 8 wave32 waves / workgroup

// addrspace(3) pointer type: pointer value is the 32-bit LDS byte offset.
typedef __attribute__((address_space(3))) float2* as3_f2_ptr;

__global__ __launch_bounds__(BLOCK) void grid_sample_bilinear_kernel(
    const float* __restrict__ x,      // (N, C, H, W)
    const float* __restrict__ grid,   // (N, GH, GW, 2)
    float* __restrict__ out)          // (N, C, GH, GW)
{
    const int tid = threadIdx.x;
    const int p   = blockIdx.x * BLOCK + tid;   // flat (n, gh, gw)

    // ---- Stage this block's 2 KB grid tile into LDS via the CDNA5 async
    // global->LDS DMA path (tracked by ASYNCcnt), then read it from LDS. ----
    __shared__ float2 sgrid[BLOCK];

    const float2* gbase = reinterpret_cast<const float2*>(grid) +
                          (size_t)blockIdx.x * BLOCK;        // uniform per block
    const uint32_t voff    = (uint32_t)tid << 3;             // per-lane byte offset
    const uint32_t lds_off = (uint32_t)(uintptr_t)((as3_f2_ptr)&sgrid[tid]);

    // GLOBAL_LOAD_ASYNC_TO_LDS_B64 (GVS mode): copies 8 bytes at saddr + vaddr to LDS byte offset vdst
    asm volatile("global_load_async_to_lds_b64 %0, %1, %2"
                 :
                 : "v"(lds_off), "v"(voff), "s"(gbase)
                 : "memory");
    asm volatile("s_wait_asynccnt 0" ::: "memory");
    __syncthreads();

    const float xg = sgrid[tid].x;
    const float yg = sgrid[tid].y;

    // Mask + weights, exactly mirroring the reference.
    const bool in = (xg >= 0.0f) && (yg >= 0.0f) &&
                    (xg < (float)(W - 1)) && (yg < (float)(H - 1));
    const float x0f = floorf(xg);
    const float y0f = floorf(yg);
    const float fx  = xg - x0f;
    const float fy  = yg - y0f;
    const float wa = (1.0f - fx) * (1.0f - fy);   // (y0, x0)
    const float wb = (1.0f - fx) * fy;            // (y1, x0)
    const float wc = fx * (1.0f - fy);            // (y0, x1)
    const float wd = fx * fy;                     // (y1, x1)
    const float mf = in ? 1.0f : 0.0f;

    // Masked-out points index (0,0) like the reference (result is zeroed).
    const int x0 = in ? (int)x0f : 0;
    const int y0 = in ? (int)y0f : 0;
    const int x1 = in ? (x0 + 1) : 0;
    const int y1 = in ? (y0 + 1) : 0;
    const int i00 = y0 * W + x0;
    const int i10 = y1 * W + x0;
    const int i01 = y0 * W + x1;
    const int i11 = y1 * W + x1;

    // Decompose flat point id (GW = GH = 512 -> shifts).
    const int n  = p >> 18;
    const int gh = (p >> 9) & 511;
    const int gw = p & 511;

    const float* xn = x + (size_t)n * C * HW;
    float* on = out + ((size_t)n * C * GH + gh) * GW + gw;  // &out[n][0][gh][gw]

    // Channel loop: 4 gathers + FMA per channel. Unroll x8 so ~32 loads are
    // in flight per thread; gathers mostly hit the 192 MB L2 (x[n] is 33.5 MB).
    // Output stores are non-temporal (th:TH_STORE_NT) to keep x resident in L2.
#pragma unroll 8
    for (int c = 0; c < C; ++c) {
        const float* xc = xn + (size_t)c * HW;
        const float v00 = xc[i00];
        const float v10 = xc[i10];
        const float v01 = xc[i01];
        const float v11 = xc[i11];
        const float v = (v00 * wa + v10 * wb) + (v01 * wc + v11 * wd);
        __builtin_nontemporal_store(v * mf, on + (size_t)c * (GH * GW));
    }
}

extern "C" void kernel_launch(void* const* d_in, const int* in_sizes, int n_in,
                              void* d_out, int out_size, void* d_ws, size_t ws_size,
                              hipStream_t stream) {
    (void)in_sizes; (void)n_in; (void)out_size; (void)d_ws; (void)ws_size;
    const float* x    = (const float*)d_in[0];   // (4, 32, 512, 512) f32
    const float* grid = (const float*)d_in[1];   // (4, 512, 512, 2)  f32
    float* out        = (float*)d_out;           // (4, 32, 512, 512) f32

    const int blocks = PTS / BLOCK;              // 4096
    grid_sample_bilinear_kernel<<<blocks, BLOCK, 0, stream>>>(x, grid, out);
}